// RGATLayer_31645319037677
// MI455X (gfx1250) — compile-verified
//
#include <hip/hip_runtime.h>
#include <hip/hip_bf16.h>

typedef float v2f __attribute__((ext_vector_type(2)));
typedef float v8f __attribute__((ext_vector_type(8)));

#define N_NODES   25000
#define N_EDGES   320000
#define IN_FEAT   64
#define OUT_FEAT  64
#define NUM_HEADS 5
#define NUM_RELS  20
#define NUM_BASES 10
#define HF        (NUM_HEADS * OUT_FEAT)   // 320
#define ATTN_ELEMS (NUM_RELS * NUM_HEADS * 2 * OUT_FEAT)  // 20*5*128 = 12800
#define MTILES ((N_NODES + 15) / 16)       // 1563

// ---------------------------------------------------------------------------
// Kernel 0: attn_fcs[r][h][f] = sum_b w_comp[r][b] * attention_weight[b][f][h]
// Stored [R][H][2F] so the edge kernel dots contiguously over f.
// ---------------------------------------------------------------------------
__global__ void compose_attn_kernel(const float* __restrict__ w_comp,        // [R][B]
                                    const float* __restrict__ attn_w,        // [B][2F][H]
                                    float* __restrict__ attn_fcs) {          // [R][H][2F]
    int t = blockIdx.x * blockDim.x + threadIdx.x;
    if (t >= ATTN_ELEMS) return;
    int r = t / (NUM_HEADS * 2 * OUT_FEAT);
    int h = (t / (2 * OUT_FEAT)) % NUM_HEADS;
    int f = t % (2 * OUT_FEAT);
    float acc = 0.f;
#pragma unroll
    for (int b = 0; b < NUM_BASES; ++b)
        acc += w_comp[r * NUM_BASES + b] * attn_w[b * (2 * OUT_FEAT * NUM_HEADS) + f * NUM_HEADS + h];
    attn_fcs[t] = acc;
}

// ---------------------------------------------------------------------------
// Kernel 1: WMMA f32 projection, Out[N][LDW] = feat[N][64] @ W[64][LDW].
// Compile-time LDW -> all weight-load offsets fold into 24-bit immediates off
// a single base pointer. Each wave owns one 16-row M-tile x NT 16-col N-tiles
// (A operand loaded once, reused NT times). K=64 via 16x V_WMMA_F32_16X16X4_F32.
//   f32 A 16x4 layout: lanes 0-15 hold K=0/1 (vgpr0/1), lanes 16-31 K=2/3.
//   f32 C/D: vgpr r -> M = r (+8 for lanes 16-31), N = lane%16.
// Loads clamp the M row (EXEC must stay all-ones for WMMA); stores are guarded.
// ---------------------------------------------------------------------------
template <int LDW, int NT>
__global__ void proj_wmma_kernel(const float* __restrict__ feat,
                                 const float* __restrict__ W,     // [64][LDW]
                                 float* __restrict__ Out) {       // [N][LDW]
    constexpr int WCOLS = LDW / (16 * NT);           // wave-columns
    const int wave = threadIdx.x >> 5;
    const int lane = threadIdx.x & 31;
    const int tile = blockIdx.x * (blockDim.x >> 5) + wave;
    if (tile >= MTILES * WCOLS) return;              // uniform per wave

    const int mt = tile / WCOLS;
    const int wc = tile % WCOLS;

    const int l16  = lane & 15;
    const int koff = (lane < 16) ? 0 : 2;            // f32 A/B K-pair select
    const int rowA = min(mt * 16 + l16, N_NODES - 1);
    const int colb = wc * (16 * NT) + l16;

    const float* ap = feat + (size_t)rowA * IN_FEAT + koff;  // A: +k imm offsets
    const float* bp = W + (size_t)koff * LDW + colb;         // B: +k*LDW+j*16 imm

    v8f c[NT];
#pragma unroll
    for (int j = 0; j < NT; ++j) c[j] = v8f{};

#pragma unroll
    for (int k = 0; k < IN_FEAT; k += 4) {
        v2f a;
        a.x = ap[k];
        a.y = ap[k + 1];
#pragma unroll
        for (int j = 0; j < NT; ++j) {
            v2f b;
            b.x = bp[k * LDW + j * 16];
            b.y = bp[(k + 1) * LDW + j * 16];
            c[j] = __builtin_amdgcn_wmma_f32_16x16x4_f32(
                       /*neg_a=*/false, a, /*neg_b=*/false, b,
                       /*c_mod=*/(short)0, c[j], /*reuse_a=*/false, /*reuse_b=*/false);
        }
    }

    const int mbase = mt * 16 + ((lane < 16) ? 0 : 8);
    float* op = Out + (size_t)mbase * LDW + colb;
#pragma unroll
    for (int r = 0; r < 8; ++r) {
        if (mbase + r < N_NODES) {
#pragma unroll
            for (int j = 0; j < NT; ++j)
                op[r * LDW + j * 16] = c[j][r];
        }
    }
}

// ---------------------------------------------------------------------------
// Kernel 2: edge attention + atomic scatter. One wave per edge (grid-stride).
// Lane l owns features 2l, 2l+1 of z_src / z_dst. attn_fcs staged in LDS
// (51.2 KB of the 320 KB WGP pool); float2 LDS reads (8B aligned).
// ---------------------------------------------------------------------------
__global__ void edge_kernel(const float* __restrict__ z,          // [N][64]
                            const int*   __restrict__ src,
                            const int*   __restrict__ dst,
                            const int*   __restrict__ etype,
                            const float* __restrict__ attn_fcs,   // [R][H][2F]
                            float*       __restrict__ out) {      // [N][320] (+=)
    __shared__ float sW[ATTN_ELEMS];
    for (int i = threadIdx.x; i < ATTN_ELEMS; i += blockDim.x) sW[i] = attn_fcs[i];
    __syncthreads();

    const int lane  = threadIdx.x & 31;
    const int wid   = blockIdx.x * (blockDim.x >> 5) + (threadIdx.x >> 5);
    const int nwave = gridDim.x * (blockDim.x >> 5);
    const int f0    = lane * 2;

    for (int e = wid; e < N_EDGES; e += nwave) {
        const int s = src[e], d = dst[e], r = etype[e];
        const float2 zs = *(const float2*)(z + (size_t)s * IN_FEAT + f0);
        const float2 zd = *(const float2*)(z + (size_t)d * IN_FEAT + f0);
        const float* Wr = sW + r * (NUM_HEADS * 2 * OUT_FEAT);

        float att[NUM_HEADS];
#pragma unroll
        for (int h = 0; h < NUM_HEADS; ++h) {
            const float* Wh = Wr + h * (2 * OUT_FEAT);
            const float2 ws = *(const float2*)(Wh + f0);
            const float2 wd = *(const float2*)(Wh + OUT_FEAT + f0);
            float p = zs.x * ws.x + zs.y * ws.y + zd.x * wd.x + zd.y * wd.y;
#pragma unroll
            for (int m = 16; m > 0; m >>= 1) p += __shfl_xor(p, m, 32);
            att[h] = (p > 0.f) ? p : 0.01f * p;      // leaky_relu; all lanes hold sum
        }

        float* op = out + (size_t)d * HF + f0;
#pragma unroll
        for (int h = 0; h < NUM_HEADS; ++h) {
            atomicAdd(op + h * OUT_FEAT,     att[h] * zs.x);
            atomicAdd(op + h * OUT_FEAT + 1, att[h] * zs.y);
        }
    }
}

// ---------------------------------------------------------------------------
extern "C" void kernel_launch(void* const* d_in, const int* in_sizes, int n_in,
                              void* d_out, int out_size, void* d_ws, size_t ws_size,
                              hipStream_t stream) {
    const float* feat      = (const float*)d_in[0];
    const int*   src       = (const int*)  d_in[1];
    const int*   dst       = (const int*)  d_in[2];
    const int*   etype     = (const int*)  d_in[3];
    const float* fc_w      = (const float*)d_in[4];
    const float* self_fc_w = (const float*)d_in[5];
    const float* attn_w    = (const float*)d_in[6];
    const float* w_comp    = (const float*)d_in[7];
    float* out = (float*)d_out;

    float* z        = (float*)d_ws;                          // 25000*64 f32 = 6.4 MB
    float* attn_fcs = z + (size_t)N_NODES * OUT_FEAT;        // 12800 f32

    // 0) compose per-relation attention matrices
    compose_attn_kernel<<<(ATTN_ELEMS + 255) / 256, 256, 0, stream>>>(w_comp, attn_w, attn_fcs);

    // 1) WMMA projections: z -> ws, self_z -> d_out (initializes accumulator)
    {
        // z = feat @ fc_w : 64 cols, 2 N-tiles per wave -> 2 wave-cols
        int tiles  = MTILES * (OUT_FEAT / 32);
        int blocks = (tiles + 3) / 4;
        proj_wmma_kernel<OUT_FEAT, 2><<<blocks, 128, 0, stream>>>(feat, fc_w, z);
    }
    {
        // self_z = feat @ self_fc_w : 320 cols, 2 N-tiles per wave -> 10 wave-cols
        int tiles  = MTILES * (HF / 32);
        int blocks = (tiles + 3) / 4;
        proj_wmma_kernel<HF, 2><<<blocks, 128, 0, stream>>>(feat, self_fc_w, out);
    }

    // 2) edge attention + scatter-add into d_out
    edge_kernel<<<1024, 256, 0, stream>>>(z, src, dst, etype, attn_fcs, out);
}